// QuantumConvolutionLayer_26989574488828
// MI455X (gfx1250) — compile-verified
//
#include <hip/hip_runtime.h>
#include <hip/hip_bf16.h>
#include <stdint.h>

// -------- complex helpers for the one-time circuit collapse --------
struct Cx { float re, im; };
__device__ inline Cx cmul(Cx a, Cx b) { return {a.re*b.re - a.im*b.im, a.re*b.im + a.im*b.re}; }
__device__ inline Cx cadd(Cx a, Cx b) { return {a.re + b.re, a.im + b.im}; }

__device__ inline void mm4(const Cx A[4][4], const Cx B[4][4], Cx O[4][4]) {
    for (int i = 0; i < 4; ++i)
        for (int j = 0; j < 4; ++j) {
            Cx acc{0.f, 0.f};
            for (int k = 0; k < 4; ++k) acc = cadd(acc, cmul(A[i][k], B[k][j]));
            O[i][j] = acc;
        }
}

// Build the two 4x4 real symmetric quadratic-form matrices M0, M1 from weights.
// U = g6 g5 g4 g3 g2 g1 ;  zc = psi0^T Mc psi0 with
// Mc[i][j] = sum_k sgn_c[k] * (ReU[k][i]ReU[k][j] + ImU[k][i]ImU[k][j])
// Weights are in [0, 4pi) -> fast __sincosf is plenty accurate and avoids
// dragging in libm's Payne-Hanek huge-argument reduction (~1.4k SALU ops).
__global__ void qconv_build_quadforms(const float* __restrict__ w, float* __restrict__ M)
{
    if (threadIdx.x != 0 || blockIdx.x != 0) return;

    Cx RX0[2][2], RY[2][2], RZ[2][2], RX1[2][2];
    { float s, c; __sincosf(0.5f * w[0], &s, &c);
      RX0[0][0]={c,0.f}; RX0[0][1]={0.f,-s}; RX0[1][0]={0.f,-s}; RX0[1][1]={c,0.f}; }
    { float s, c; __sincosf(0.5f * w[1], &s, &c);
      RY[0][0]={c,0.f};  RY[0][1]={-s,0.f};  RY[1][0]={s,0.f};   RY[1][1]={c,0.f}; }
    { float s, c; __sincosf(0.5f * w[2], &s, &c);
      RZ[0][0]={c,-s};   RZ[0][1]={0.f,0.f}; RZ[1][0]={0.f,0.f}; RZ[1][1]={c,s}; }
    { float s, c; __sincosf(0.5f * w[3], &s, &c);
      RX1[0][0]={c,0.f}; RX1[0][1]={0.f,-s}; RX1[1][0]={0.f,-s}; RX1[1][1]={c,0.f}; }

    Cx U[4][4], G[4][4], T[4][4];

    // g1 = kron(I, RX0)
    for (int a=0;a<2;++a) for (int ap=0;ap<2;++ap)
      for (int b2=0;b2<2;++b2) for (int bp=0;bp<2;++bp)
        U[2*a+b2][2*ap+bp] = (a==ap) ? RX0[b2][bp] : Cx{0.f,0.f};
    // g2 = kron(RY, I)
    for (int a=0;a<2;++a) for (int ap=0;ap<2;++ap)
      for (int b2=0;b2<2;++b2) for (int bp=0;bp<2;++bp)
        G[2*a+b2][2*ap+bp] = (b2==bp) ? RY[a][ap] : Cx{0.f,0.f};
    mm4(G, U, T);                                   // T = g2 g1
    // g3 = CNOT10 (row permutation {0,3,2,1})
    { const int f3[4] = {0,3,2,1};
      for (int i=0;i<4;++i) for (int j=0;j<4;++j) U[i][j] = T[f3[i]][j]; }   // U = g3 g2 g1
    // g4 = CNOT01 (row permutation {0,1,3,2})
    { const int f4[4] = {0,1,3,2};
      for (int i=0;i<4;++i) for (int j=0;j<4;++j) T[i][j] = U[f4[i]][j]; }   // T = g4 g3 g2 g1
    // g5 = kron(I, RZ)
    for (int a=0;a<2;++a) for (int ap=0;ap<2;++ap)
      for (int b2=0;b2<2;++b2) for (int bp=0;bp<2;++bp)
        G[2*a+b2][2*ap+bp] = (a==ap) ? RZ[b2][bp] : Cx{0.f,0.f};
    mm4(G, T, U);                                   // U = g5 g4 g3 g2 g1
    // g6 = kron(RX1, I)
    for (int a=0;a<2;++a) for (int ap=0;ap<2;++ap)
      for (int b2=0;b2<2;++b2) for (int bp=0;bp<2;++bp)
        G[2*a+b2][2*ap+bp] = (b2==bp) ? RX1[a][ap] : Cx{0.f,0.f};
    mm4(G, U, T);                                   // T = full circuit unitary

    const float sg0[4] = {1.f, 1.f, -1.f, -1.f};
    const float sg1[4] = {1.f, -1.f, 1.f, -1.f};
    for (int i=0;i<4;++i)
        for (int j=0;j<4;++j) {
            float m0 = 0.f, m1 = 0.f;
            for (int k=0;k<4;++k) {
                float rr = T[k][i].re*T[k][j].re + T[k][i].im*T[k][j].im;
                m0 += sg0[k]*rr;
                m1 += sg1[k]*rr;
            }
            M[i*4 + j]      = m0;
            M[16 + i*4 + j] = m1;
        }
}

// -------- main memory-bound kernel (HBM roofline ~2.9 us) --------
// x: (1024, 128, 128) f32 ; out: (1024, 64, 64, 2) f32
// Each block: 4 "rows" (b,p). CDNA5 async global->LDS stages 4 x 512B input
// rows (no VGPR round-trip), then 256 threads each produce one (q, z0, z1).
typedef __attribute__((address_space(3))) float lds_float;

__global__ __launch_bounds__(256) void qconv_main(const float* __restrict__ x,
                                                  const float* __restrict__ M,
                                                  float* __restrict__ out)
{
    __shared__ float smem[4 * 128];   // 4 rows x 128 floats = 2 KB

    const int tid = threadIdx.x;
    const int blk = blockIdx.x;

    // ---- async global -> LDS staging (128 lanes x 16B = 2 KB) ----
    if (tid < 128) {
        const int rl = tid >> 5;          // local row 0..3
        const int ck = tid & 31;          // 16-byte chunk 0..31
        const int r  = blk * 4 + rl;      // global row index = b*64 + p
        const int b  = r >> 6;
        const int p  = r & 63;
        int j = 2 * p + 1; if (j > 126) j = 126;          // row clamp (p == 63)
        const uint32_t voff = ((uint32_t)(b * 16384 + j * 128) << 2) + ((uint32_t)ck << 4);
        const uint32_t lds  = (uint32_t)(size_t)(lds_float*)smem
                            + (uint32_t)(rl * 512 + (ck << 4));
        asm volatile("global_load_async_to_lds_b128 %0, %1, %2"
                     :: "v"(lds), "v"(voff), "s"(x)
                     : "memory");
    }
    asm volatile("s_wait_asynccnt 0" ::: "memory");
    __syncthreads();

    // ---- compute: one output pixel per thread ----
    const int rl = tid >> 6;              // local row 0..3
    const int q  = tid & 63;
    int k = 2 * q + 1; if (k > 126) k = 126;              // column clamp (q == 63)

    const float phi0 = smem[rl * 128 + k];
    const float phi1 = smem[rl * 128 + k + 1];

    float s0, c0, s1, c1;
    __sincosf(1.57079632679f * phi0, &s0, &c0);
    __sincosf(1.57079632679f * phi1, &s1, &c1);

    float ps[4];
    ps[0] = c0 * c1; ps[1] = c0 * s1; ps[2] = s0 * c1; ps[3] = s0 * s1;

    float z0 = 0.f, z1 = 0.f;
#pragma unroll
    for (int i = 0; i < 4; ++i) {
        const float t0 = M[4*i+0]*ps[0] + M[4*i+1]*ps[1] + M[4*i+2]*ps[2] + M[4*i+3]*ps[3];
        const float t1 = M[16+4*i+0]*ps[0] + M[16+4*i+1]*ps[1] + M[16+4*i+2]*ps[2] + M[16+4*i+3]*ps[3];
        z0 += ps[i] * t0;
        z1 += ps[i] * t1;
    }

    const int r = blk * 4 + rl;
    float2* o = (float2*)out;
    o[(size_t)r * 64 + q] = make_float2(z0, z1);          // (.., q, {z0,z1}) layout
}

extern "C" void kernel_launch(void* const* d_in, const int* in_sizes, int n_in,
                              void* d_out, int out_size, void* d_ws, size_t ws_size,
                              hipStream_t stream)
{
    const float* x = (const float*)d_in[0];   // (1024,128,128) f32
    const float* w = (const float*)d_in[1];   // (2,2) f32
    float* M = (float*)d_ws;                  // 32 floats: M0 | M1

    qconv_build_quadforms<<<1, 32, 0, stream>>>(w, M);

    // 1024*64 = 65536 rows, 4 rows/block -> 16384 blocks of 256 threads
    qconv_main<<<16384, 256, 0, stream>>>(x, M, (float*)d_out);
}